// NGramEmbedding_73718818668652
// MI455X (gfx1250) — compile-verified
//
#include <hip/hip_runtime.h>

// NGram embedding for gfx1250 (MI455X), two-phase:
//  Phase A: thread-per-position id computation. Branch-free clamped window
//           loads (one clause, one wait) + vectorized b128/b64 id stores.
//  Phase B: wave32-per-position gather-accumulate; ids fetched as wave-uniform
//           b128 broadcasts -> v_readfirstlane -> SADDR-form global_load_b128
//           gathers; NT b128 stores keep the 75.5MB tables resident in L2.
// Fallback: fused kernel if d_ws is too small (decision depends only on
// ws_size -> deterministic / graph-capture safe).

typedef __attribute__((ext_vector_type(4))) float v4f;
typedef __attribute__((ext_vector_type(4))) int   v4i;
typedef __attribute__((ext_vector_type(2))) int   v2i;

#define N_TBL    4096
#define MIXED_ID 4096
#define N_SIZES  18
#define MAXN     20
#define IDS_PITCH 20          // ints per position in ws (16B-aligned rows)
#define SEQ_L    8192
#define BATCH    8
#define DIM      256

// ---------------- Phase A: one thread per position ----------------
__global__ __launch_bounds__(256) void ngram_ids_kernel(
    const int* __restrict__ byte_ids,
    int* __restrict__ ids_out) {
  const int p = blockIdx.x * 256 + threadIdx.x;   // position in [0, B*L)
  const int i = p & (SEQ_L - 1);
  const int rowbase = p - i;

  // Branch-free backward window: offset j reads byte_ids[rowbase + max(i-j,0)].
  // Always in-bounds -> 20 unconditional coalesced loads (single wait). The
  // clamped value only feeds hashes whose window is already flagged bad.
  int xv[MAXN];
#pragma unroll
  for (int j = 0; j < MAXN; ++j) {
    const int jj = i - j;
    xv[j] = byte_ids[rowbase + (jj > 0 ? jj : 0)];
  }

  unsigned badbits = 0u;      // bit j set if backward offset j is bad
  unsigned s[MAXN];
#pragma unroll
  for (int j = 0; j < MAXN; ++j) {
    const int d = xv[j] - 1;                        // DNA iff byte in [1,4]
    const bool bd = (i - j < 0) | ((unsigned)d >= 4u);
    s[j] = bd ? 0u : (unsigned)d;
    badbits |= bd ? (1u << j) : 0u;
  }

  // h_k = sum_{j=0}^{k-1} s[j] * 31^j (mod 4096)
  int idv[N_SIZES];
  unsigned h  = s[0];
  unsigned pw = 1u;
#pragma unroll
  for (int j = 1; j < MAXN; ++j) {
    pw = (pw * 31u) & (unsigned)(N_TBL - 1);
    h  = (h + s[j] * pw) & (unsigned)(N_TBL - 1);
    const int k = j + 1;
    if (k >= 3) {
      const unsigned wm = (1u << k) - 1u;           // offsets 0..k-1
      idv[k - 3] = (badbits & wm) ? MIXED_ID : (int)h;
    }
  }

  // Vectorized stores: 4 x b128 + 1 x b64 (rows are 16B aligned: pitch 80B).
  v4i* __restrict__ op = (v4i*)(ids_out + (size_t)p * IDS_PITCH);
  const v4i w0 = {idv[0],  idv[1],  idv[2],  idv[3]};
  const v4i w1 = {idv[4],  idv[5],  idv[6],  idv[7]};
  const v4i w2 = {idv[8],  idv[9],  idv[10], idv[11]};
  const v4i w3 = {idv[12], idv[13], idv[14], idv[15]};
  const v2i w4 = {idv[16], idv[17]};
  op[0] = w0;
  op[1] = w1;
  op[2] = w2;
  op[3] = w3;
  *(v2i*)(op + 4) = w4;
}

// ---------------- Phase B: wave32 per position ----------------
__global__ __launch_bounds__(256) void ngram_gather_kernel(
    const int* __restrict__ ids_in,
    const float* __restrict__ tables,
    float* __restrict__ out) {
  const int lane = threadIdx.x & 31;
  const int p    = blockIdx.x * 8 + (threadIdx.x >> 5);

  // Wave-uniform id fetch: all lanes load the same 16B-aligned 80B row
  // (hardware broadcast), then pin each id into an SGPR.
  const v4i* idp = (const v4i*)(ids_in + (size_t)p * IDS_PITCH);
  const v4i q0 = idp[0];
  const v4i q1 = idp[1];
  const v4i q2 = idp[2];
  const v4i q3 = idp[3];
  const v2i q4 = *(const v2i*)(idp + 4);
  const int ids[N_SIZES] = {q0.x, q0.y, q0.z, q0.w,
                            q1.x, q1.y, q1.z, q1.w,
                            q2.x, q2.y, q2.z, q2.w,
                            q3.x, q3.y, q3.z, q3.w,
                            q4.x, q4.y};

  const int doff = lane * 8;                        // 8 floats per lane
  v4f acc0 = {0.f, 0.f, 0.f, 0.f};
  v4f acc1 = {0.f, 0.f, 0.f, 0.f};
#pragma unroll
  for (int t = 0; t < N_SIZES; ++t) {
    const int id = __builtin_amdgcn_readfirstlane(ids[t]); // -> SADDR gathers
    const float* row =
        tables + ((size_t)t * (size_t)(N_TBL + 1) + (size_t)id) * (size_t)DIM;
    acc0 += *(const v4f*)(row + doff);
    acc1 += *(const v4f*)(row + doff + 4);
  }
  acc0 /= 19.0f;                                    // match reference rounding
  acc1 /= 19.0f;

  float* orow = out + (size_t)p * (size_t)DIM + doff;
  __builtin_nontemporal_store(acc0, (v4f*)orow);    // streaming output: keep
  __builtin_nontemporal_store(acc1, (v4f*)(orow + 4)); // L2 for the tables
}

// ---------------- Fallback: fused (used only if ws too small) ----------------
__global__ __launch_bounds__(256) void ngram_fused_kernel(
    const int* __restrict__ byte_ids,
    const float* __restrict__ tables,
    float* __restrict__ out) {
  const int lane = threadIdx.x & 31;
  const int p    = blockIdx.x * 8 + (threadIdx.x >> 5);
  const int i    = p & (SEQ_L - 1);
  const int rowbase = p - i;

  // Branch-free: every lane loads a clamped in-bounds address; only lanes
  // 0..19 (window offsets) can be marked bad / are read by the shuffles.
  const int jpos = i - (MAXN - 1) + lane;           // lane l -> offset 19-l
  const int cl   = jpos < 0 ? 0 : (jpos > (SEQ_L - 1) ? (SEQ_L - 1) : jpos);
  const int d    = byte_ids[rowbase + cl] - 1;
  const bool inw = (lane < MAXN);
  const bool bad = inw & ((jpos < 0) | ((unsigned)d >= 4u));
  const int safe = ((unsigned)d < 4u) ? d : 0;
  const unsigned badmask = (unsigned)__ballot(bad);

  int ids[N_SIZES];
  unsigned h  = (unsigned)__shfl(safe, MAXN - 1);
  unsigned pw = 1u;
#pragma unroll
  for (int j = 1; j < MAXN; ++j) {
    const int sj = __shfl(safe, MAXN - 1 - j);
    pw = (pw * 31u) & (unsigned)(N_TBL - 1);
    h  = (h + (unsigned)sj * pw) & (unsigned)(N_TBL - 1);
    const int k = j + 1;
    if (k >= 3) {
      const unsigned wmask = ((1u << k) - 1u) << (MAXN - k);
      ids[k - 3] = (badmask & wmask) ? MIXED_ID : (int)h;
    }
  }

  const int doff = lane * 8;
  v4f acc0 = {0.f, 0.f, 0.f, 0.f};
  v4f acc1 = {0.f, 0.f, 0.f, 0.f};
#pragma unroll
  for (int t = 0; t < N_SIZES; ++t) {
    const int id = __builtin_amdgcn_readfirstlane(ids[t]);
    const float* row =
        tables + ((size_t)t * (size_t)(N_TBL + 1) + (size_t)id) * (size_t)DIM;
    acc0 += *(const v4f*)(row + doff);
    acc1 += *(const v4f*)(row + doff + 4);
  }
  acc0 /= 19.0f;
  acc1 /= 19.0f;

  float* orow = out + (size_t)p * (size_t)DIM + doff;
  __builtin_nontemporal_store(acc0, (v4f*)orow);
  __builtin_nontemporal_store(acc1, (v4f*)(orow + 4));
}

extern "C" void kernel_launch(void* const* d_in, const int* in_sizes, int n_in,
                              void* d_out, int out_size, void* d_ws, size_t ws_size,
                              hipStream_t stream) {
  (void)in_sizes; (void)n_in; (void)out_size;
  const int*   byte_ids = (const int*)d_in[0];    // [8, 8192] int32
  const float* tables   = (const float*)d_in[1];  // [18, 4097, 256] fp32
  float*       out      = (float*)d_out;          // [8, 8192, 256] fp32

  const int positions = BATCH * SEQ_L;            // 65536
  const size_t ws_need = (size_t)positions * IDS_PITCH * sizeof(int); // 5.24 MB

  if (d_ws != nullptr && ws_size >= ws_need) {
    int* ids_ws = (int*)d_ws;
    ngram_ids_kernel<<<dim3(positions / 256), dim3(256), 0, stream>>>(
        byte_ids, ids_ws);
    ngram_gather_kernel<<<dim3(positions / 8), dim3(256), 0, stream>>>(
        ids_ws, tables, out);
  } else {
    ngram_fused_kernel<<<dim3(positions / 8), dim3(256), 0, stream>>>(
        byte_ids, tables, out);
  }
}